// AdaptiveScaleDecoupledMamba_84026740179569
// MI455X (gfx1250) — compile-verified
//
#include <hip/hip_runtime.h>
#include <hip/hip_bf16.h>
#include <math.h>

// ---------------- problem constants ----------------
#define NB   16
#define LSEQ 512
#define DM   512
#define DI   1024
#define DS   16
#define DCONV 4
#define DTR  32
#define ROWS (NB * LSEQ)          // 8192 token rows

// epilogue modes for the generic WMMA GEMM
#define EP_NONE     0
#define EP_SOFTPLUS 1
#define EP_GELU     2
#define EP_BIAS     3

typedef __attribute__((ext_vector_type(2))) float v2f;
typedef __attribute__((ext_vector_type(8))) float v8f;

__device__ __forceinline__ float gelu_exact(float x) {
    return 0.5f * x * (1.0f + erff(x * 0.70710678118654752440f));
}
__device__ __forceinline__ float softplus_f(float x) {
    return (x > 20.0f) ? x : log1pf(expf(x));
}
__device__ __forceinline__ float silu_f(float x) {
    return x / (1.0f + expf(-x));
}

// ---------------------------------------------------------------------------
// Generic NT GEMM with f32 WMMA:  C[M,N] = A[M,K] * B[N,K]^T  (+epilogue)
// One wave computes a (16*MT)x64 strip of C (MT x 4 WMMA tiles).
//  - B fragments shared across MT row tiles (less load traffic per FLOP)
//  - software-pipelined: next k-step's fragments are loaded before the
//    current k-step's WMMAs are issued, so waits overlap the matrix pipe.
// f32 WMMA 16x16x4 fragment layout (ISA 7.12.2):
//   A: lane l holds A[l%16, k0 + 2*(l/16) + {0,1}]   -> one float2 load
//   B: lane l holds B[k0 + 2*(l/16) + {0,1}, l%16]   -> float2 from W row
//   C: vgpr r, lane l -> C[r + 8*(l/16), l%16]
// ---------------------------------------------------------------------------
__device__ __forceinline__ void store_tile(v8f acc, float* __restrict__ C,
                                           const float* __restrict__ bias,
                                           int rbase, int cb, int ldc, int mode) {
#pragma unroll
    for (int r = 0; r < 8; ++r) {
        float v = acc[r];
        if (mode == EP_SOFTPLUS)      v = softplus_f(v + bias[cb]);
        else if (mode == EP_GELU)     v = gelu_exact(v + bias[cb]);
        else if (mode == EP_BIAS)     v = v + bias[cb];
        C[(size_t)(rbase + r) * ldc + cb] = v;
    }
}

template <int MT>
__global__ void k_wmma_gemm_nt(const float* __restrict__ A, const float* __restrict__ B,
                               float* __restrict__ C, const float* __restrict__ bias,
                               int M, int N, int K, int lda, int ldb, int ldc, int mode) {
    const int lane   = threadIdx.x & 31;
    const int wave   = threadIdx.x >> 5;
    const int ntiles = N >> 6;                          // 64 columns per wave
    const int wid    = blockIdx.x * 8 + wave;
    const int total  = (M / (16 * MT)) * ntiles;
    if (wid >= total) return;                           // wave-uniform: EXEC stays full

    const int tg = wid / ntiles, tn = wid % ntiles;
    const int row0 = tg * 16 * MT, col0 = tn << 6;
    const int l16 = lane & 15;
    const int kh  = (lane >> 4) << 1;                   // 0 or 2

    const float* Ap[MT];
#pragma unroll
    for (int m = 0; m < MT; ++m)
        Ap[m] = A + (size_t)(row0 + 16 * m + l16) * lda + kh;
    const float* Bp[4];
#pragma unroll
    for (int j = 0; j < 4; ++j)
        Bp[j] = B + (size_t)(col0 + 16 * j + l16) * ldb + kh;

    v8f acc[MT][4];
    const v8f vzero = {};
#pragma unroll
    for (int m = 0; m < MT; ++m)
#pragma unroll
        for (int j = 0; j < 4; ++j) acc[m][j] = vzero;

    // pipeline prologue: fragments for k = 0
    v2f a[MT], b[4];
#pragma unroll
    for (int m = 0; m < MT; ++m) a[m] = *(const v2f*)(Ap[m]);
#pragma unroll
    for (int j = 0; j < 4; ++j) b[j] = *(const v2f*)(Bp[j]);

    for (int k = 0; k < K - 4; k += 4) {
        // issue next k-step's loads before consuming current fragments
        v2f an[MT], bn[4];
#pragma unroll
        for (int m = 0; m < MT; ++m) an[m] = *(const v2f*)(Ap[m] + k + 4);
#pragma unroll
        for (int j = 0; j < 4; ++j) bn[j] = *(const v2f*)(Bp[j] + k + 4);
#pragma unroll
        for (int m = 0; m < MT; ++m)
#pragma unroll
            for (int j = 0; j < 4; ++j)
                acc[m][j] = __builtin_amdgcn_wmma_f32_16x16x4_f32(
                    false, a[m], false, b[j], (short)0, acc[m][j], false, false);
#pragma unroll
        for (int m = 0; m < MT; ++m) a[m] = an[m];
#pragma unroll
        for (int j = 0; j < 4; ++j) b[j] = bn[j];
    }
    // pipeline epilogue: last k-step
#pragma unroll
    for (int m = 0; m < MT; ++m)
#pragma unroll
        for (int j = 0; j < 4; ++j)
            acc[m][j] = __builtin_amdgcn_wmma_f32_16x16x4_f32(
                false, a[m], false, b[j], (short)0, acc[m][j], false, false);

    // store with fused epilogue
#pragma unroll
    for (int m = 0; m < MT; ++m) {
        const int rbase = row0 + 16 * m + ((lane >> 4) << 3);
#pragma unroll
        for (int j = 0; j < 4; ++j)
            store_tile(acc[m][j], C, bias, rbase, col0 + 16 * j + l16, ldc, mode);
    }
}

// ---------------- mean over sequence: xmean[n,d] = mean_t x[n,t,d] -----------
__global__ void k_mean(const float* __restrict__ x, float* __restrict__ xmean) {
    int id = blockIdx.x * blockDim.x + threadIdx.x;
    if (id >= NB * DM) return;
    int n = id / DM, d = id % DM;
    float s = 0.0f;
    for (int t = 0; t < LSEQ; ++t) s += x[(size_t)(n * LSEQ + t) * DM + d];
    xmean[id] = s * (1.0f / (float)LSEQ);
}

// ---------------- fill cond[:, 128:256] with emb[scale_id] ------------------
__global__ void k_embfill(const float* __restrict__ emb, const int* __restrict__ scale_id,
                          float* __restrict__ cond) {
    int id = blockIdx.x * blockDim.x + threadIdx.x;
    if (id >= NB * 128) return;
    int n = id / 128, j = id % 128;
    cond[n * 256 + 128 + j] = emb[scale_id[0] * 128 + j];
}

// ---------------- LayerNorm of dA/dB/dC segments of off ---------------------
// block per (n, seg): seg 0=dA (scaled by sc), 1=dB, 2=dC
__global__ void k_ln(const float* __restrict__ off, const float* __restrict__ g,
                     const float* __restrict__ b, const int* __restrict__ scale_id,
                     float* __restrict__ dabc) {
    __shared__ float redS[8], redQ[8];
    __shared__ float mu_s, rstd_s;
    int blk = blockIdx.x;                 // 0..47
    int n = blk / 3, seg = blk % 3;
    const float* row = off + n * 1536 + seg * DM;
    int tid = threadIdx.x;
    float v0 = row[tid], v1 = row[tid + 256];
    float s = v0 + v1, q = v0 * v0 + v1 * v1;
#pragma unroll
    for (int o = 16; o > 0; o >>= 1) {
        s += __shfl_down(s, o, 32);
        q += __shfl_down(q, o, 32);
    }
    if ((tid & 31) == 0) { redS[tid >> 5] = s; redQ[tid >> 5] = q; }
    __syncthreads();
    if (tid == 0) {
        float ts = 0.0f, tq = 0.0f;
        for (int i = 0; i < 8; ++i) { ts += redS[i]; tq += redQ[i]; }
        float mu = ts / (float)DM;
        mu_s = mu;
        rstd_s = rsqrtf(tq / (float)DM - mu * mu + 1e-5f);
    }
    __syncthreads();
    float sc = (seg == 0) ? (0.9f - 0.3f * (float)scale_id[0]) : 1.0f;
    float o0 = ((v0 - mu_s) * rstd_s * g[tid] + b[tid]) * sc;
    float o1 = ((v1 - mu_s) * rstd_s * g[tid + 256] + b[tid + 256]) * sc;
    dabc[(seg * NB + n) * DM + tid]       = o0;
    dabc[(seg * NB + n) * DM + tid + 256] = o1;
}

// ---------------- residual = x + dB (+h_init at t=0); hn = rmsnorm ----------
__global__ void k_resid_rms(const float* __restrict__ x, const float* __restrict__ dabc,
                            const float* __restrict__ h_init, const float* __restrict__ rms_w,
                            float* __restrict__ resid_out, float* __restrict__ hn) {
    __shared__ float red[8];
    __shared__ float rms_s;
    int r = blockIdx.x;                   // token row
    int n = r / LSEQ, t = r % LSEQ;
    int tid = threadIdx.x;
    const float* dB = dabc + (1 * NB + n) * DM;
    float a0 = (t == 0) ? h_init[n * DM + tid]       : 0.0f;
    float a1 = (t == 0) ? h_init[n * DM + tid + 256] : 0.0f;
    float v0 = x[(size_t)r * DM + tid]       + dB[tid]       + a0;
    float v1 = x[(size_t)r * DM + tid + 256] + dB[tid + 256] + a1;
    resid_out[(size_t)r * DM + tid]       = v0;
    resid_out[(size_t)r * DM + tid + 256] = v1;
    float q = v0 * v0 + v1 * v1;
#pragma unroll
    for (int o = 16; o > 0; o >>= 1) q += __shfl_down(q, o, 32);
    if ((tid & 31) == 0) red[tid >> 5] = q;
    __syncthreads();
    if (tid == 0) {
        float tq = 0.0f;
        for (int i = 0; i < 8; ++i) tq += red[i];
        rms_s = rsqrtf(tq / (float)DM + 1e-5f);
    }
    __syncthreads();
    hn[(size_t)r * DM + tid]       = v0 * rms_s * rms_w[tid];
    hn[(size_t)r * DM + tid + 256] = v1 * rms_s * rms_w[tid + 256];
}

// ---------------- causal depthwise conv (4 taps) + SiLU ---------------------
// backward==1 operates in flipped time (reads xs at L-1-tt), output stays in
// flipped index space for the backward scan.
__global__ void k_conv_silu(const float* __restrict__ xz, const float* __restrict__ w,
                            const float* __restrict__ b, float* __restrict__ xc,
                            int backward) {
    int id = blockIdx.x * blockDim.x + threadIdx.x;
    if (id >= ROWS * DI) return;
    int d = id % DI;
    int r = id / DI;
    int n = r / LSEQ, t = r % LSEQ;
    float acc = b[d];
#pragma unroll
    for (int j = 0; j < DCONV; ++j) {
        int tt = t - (DCONV - 1) + j;
        if (tt < 0) continue;
        int torig = backward ? (LSEQ - 1 - tt) : tt;
        acc += w[d * DCONV + j] * xz[(size_t)(n * LSEQ + torig) * (2 * DI) + d];
    }
    xc[(size_t)r * DI + d] = silu_f(acc);
}

// ---------------- selective scan: one lane per (n, d), 16 states ------------
// forward:  y[n,t,d]        = scan + u*D   (written in place over xc buffer)
// backward: y[n,L-1-t,d]   += scan + u*D   (accumulated into forward y buffer)
__global__ void k_scan(const float* __restrict__ xc, const float* __restrict__ dt,
                       const float* __restrict__ dbl, const float* __restrict__ Alog,
                       const float* __restrict__ Dp, float* __restrict__ y, int backward) {
    int g = blockIdx.x * blockDim.x + threadIdx.x;
    if (g >= NB * DI) return;
    int n = g / DI, d = g % DI;
    float A[DS], h[DS];
#pragma unroll
    for (int s = 0; s < DS; ++s) { A[s] = -expf(Alog[d * DS + s]); h[s] = 0.0f; }
    float Dd = Dp[d];
    for (int t = 0; t < LSEQ; ++t) {
        size_t rb = (size_t)(n * LSEQ + t);
        float dtv = dt[rb * DI + d];
        float u   = xc[rb * DI + d];
        float du  = dtv * u;
        const float* bc = dbl + rb * 64;   // wave-uniform address (shared n,t)
        float yv = 0.0f;
#pragma unroll
        for (int s = 0; s < DS; ++s) {
            float Bs = bc[DTR + s];
            float Cs = bc[DTR + DS + s];
            h[s] = expf(dtv * A[s]) * h[s] + du * Bs;
            yv += h[s] * Cs;
        }
        yv += u * Dd;
        if (backward) {
            size_t ob = (size_t)(n * LSEQ + (LSEQ - 1 - t));
            y[ob * DI + d] += yv;
        } else {
            y[rb * DI + d] = yv;
        }
    }
}

// ---------------- gate: ysum *= silu(z) ------------------------------------
__global__ void k_gate(float* __restrict__ ysum, const float* __restrict__ xz) {
    int id = blockIdx.x * blockDim.x + threadIdx.x;
    if (id >= ROWS * DI) return;
    int d = id % DI;
    int r = id / DI;
    float z = xz[(size_t)r * (2 * DI) + DI + d];
    ysum[id] *= silu_f(z);
}

// ---------------- final: out = lin*(1+0.1dA) + 0.1dC + res_w*resid ----------
__global__ void k_final(const float* __restrict__ outlin, const float* __restrict__ dabc,
                        const float* __restrict__ res_w, const float* __restrict__ resid,
                        float* __restrict__ out, float* __restrict__ hfinal) {
    int id = blockIdx.x * blockDim.x + threadIdx.x;
    if (id >= ROWS * DM) return;
    int c = id % DM;
    int r = id / DM;
    int n = r / LSEQ, t = r % LSEQ;
    float dA = dabc[(0 * NB + n) * DM + c];
    float dC = dabc[(2 * NB + n) * DM + c];
    float v = outlin[id] * (1.0f + 0.1f * dA) + 0.1f * dC + res_w[0] * resid[id];
    out[id] = v;
    if (t == LSEQ - 1) hfinal[n * DM + c] = v;
}

// ---------------------------------------------------------------------------
extern "C" void kernel_launch(void* const* d_in, const int* in_sizes, int n_in,
                              void* d_out, int out_size, void* d_ws, size_t ws_size,
                              hipStream_t stream) {
    const float* x        = (const float*)d_in[0];
    const int*   scale_id = (const int*)  d_in[1];
    const float* h_init   = (const float*)d_in[2];
    const float* W_gf     = (const float*)d_in[3];
    const float* b_gf     = (const float*)d_in[4];
    const float* emb      = (const float*)d_in[5];
    const float* W1       = (const float*)d_in[6];
    const float* b1       = (const float*)d_in[7];
    const float* W2       = (const float*)d_in[8];
    const float* b2       = (const float*)d_in[9];
    const float* W3       = (const float*)d_in[10];
    const float* b3       = (const float*)d_in[11];
    const float* ln_g     = (const float*)d_in[12];
    const float* ln_b     = (const float*)d_in[13];
    const float* rms_w    = (const float*)d_in[14];
    const float* W_in     = (const float*)d_in[15];
    const float* W_out    = (const float*)d_in[16];
    const float* res_w    = (const float*)d_in[17];
    const float* convw_f  = (const float*)d_in[18];
    const float* convb_f  = (const float*)d_in[19];
    const float* xproj_f  = (const float*)d_in[20];
    const float* dtw_f    = (const float*)d_in[21];
    const float* dtb_f    = (const float*)d_in[22];
    const float* Alog_f   = (const float*)d_in[23];
    const float* D_f      = (const float*)d_in[24];
    const float* convw_b  = (const float*)d_in[25];
    const float* convb_b  = (const float*)d_in[26];
    const float* xproj_b  = (const float*)d_in[27];
    const float* dtw_b    = (const float*)d_in[28];
    const float* dtb_b    = (const float*)d_in[29];
    const float* Alog_b   = (const float*)d_in[30];
    const float* D_b      = (const float*)d_in[31];

    float* outp   = (float*)d_out;                       // (N,L,DM)
    float* residp = outp + (size_t)ROWS * DM;            // (N,L,DM)
    float* hfinp  = residp + (size_t)ROWS * DM;          // (N,DM)

    // workspace layout (floats)
    float* ws = (float*)d_ws;
    size_t o = 0;
    float* xmean = ws + o; o += NB * DM;
    float* cond  = ws + o; o += NB * 256;
    float* h1    = ws + o; o += NB * DM;
    float* h2    = ws + o; o += NB * DM;
    float* off   = ws + o; o += NB * 1536;
    float* dabc  = ws + o; o += 3 * NB * DM;
    float* hn    = ws + o; o += (size_t)ROWS * DM;       // later reused for outlin
    float* xzb   = ws + o; o += (size_t)ROWS * 2 * DI;
    float* xcf   = ws + o; o += (size_t)ROWS * DI;       // later reused as ysum
    float* xcb   = ws + o; o += (size_t)ROWS * DI;
    float* dblf  = ws + o; o += (size_t)ROWS * 64;
    float* dblb  = ws + o; o += (size_t)ROWS * 64;
    float* dtf   = ws + o; o += (size_t)ROWS * DI;
    float* dtbk  = ws + o; o += (size_t)ROWS * DI;
    (void)ws_size; (void)n_in; (void)in_sizes; (void)out_size;

    auto gemm = [&](const float* A, const float* B, float* C, const float* bias,
                    int M, int N, int K, int lda, int ldb, int ldc, int mode) {
        if ((M % 32) == 0) {
            int tiles  = (M / 32) * (N / 64);
            int blocks = (tiles + 7) / 8;
            k_wmma_gemm_nt<2><<<blocks, 256, 0, stream>>>(A, B, C, bias, M, N, K, lda, ldb, ldc, mode);
        } else {
            int tiles  = (M / 16) * (N / 64);
            int blocks = (tiles + 7) / 8;
            k_wmma_gemm_nt<1><<<blocks, 256, 0, stream>>>(A, B, C, bias, M, N, K, lda, ldb, ldc, mode);
        }
    };

    // 1) conditioning path (tiny, M=16)
    k_mean<<<(NB * DM + 255) / 256, 256, 0, stream>>>(x, xmean);
    gemm(xmean, W_gf, cond, b_gf, 16, 128, 512, 512, 512, 256, EP_GELU);
    k_embfill<<<(NB * 128 + 255) / 256, 256, 0, stream>>>(emb, scale_id, cond);
    gemm(cond, W1, h1, b1, 16, 512, 256, 256, 256, 512, EP_GELU);
    gemm(h1, W2, h2, b2, 16, 512, 512, 512, 512, 512, EP_GELU);
    gemm(h2, W3, off, b3, 16, 1536, 512, 512, 512, 1536, EP_BIAS);
    k_ln<<<3 * NB, 256, 0, stream>>>(off, ln_g, ln_b, scale_id, dabc);

    // 2) residual + RMSNorm (residual lands directly in d_out part 2)
    k_resid_rms<<<ROWS, 256, 0, stream>>>(x, dabc, h_init, rms_w, residp, hn);

    // 3) big projection xz = hn @ W_in^T   (8192 x 512 x 2048) — WMMA
    gemm(hn, W_in, xzb, nullptr, ROWS, 2 * DI, DM, DM, DM, 2 * DI, EP_NONE);

    // 4) causal depthwise conv + SiLU, both directions
    k_conv_silu<<<(ROWS * DI + 255) / 256, 256, 0, stream>>>(xzb, convw_f, convb_f, xcf, 0);
    k_conv_silu<<<(ROWS * DI + 255) / 256, 256, 0, stream>>>(xzb, convw_b, convb_b, xcb, 1);

    // 5) x-projection (dt_rank + 2*state = 64 outputs) — WMMA
    gemm(xcf, xproj_f, dblf, nullptr, ROWS, 64, DI, DI, DI, 64, EP_NONE);
    gemm(xcb, xproj_b, dblb, nullptr, ROWS, 64, DI, DI, DI, 64, EP_NONE);

    // 6) dt = softplus(dbl[:, :32] @ dtw^T + dtb) — WMMA with fused epilogue
    gemm(dblf, dtw_f, dtf,  dtb_f, ROWS, DI, DTR, 64, DTR, DI, EP_SOFTPLUS);
    gemm(dblb, dtw_b, dtbk, dtb_b, ROWS, DI, DTR, 64, DTR, DI, EP_SOFTPLUS);

    // 7) selective scans: forward writes y in place over xcf; backward adds flipped
    k_scan<<<(NB * DI + 255) / 256, 256, 0, stream>>>(xcf, dtf,  dblf, Alog_f, D_f, xcf, 0);
    k_scan<<<(NB * DI + 255) / 256, 256, 0, stream>>>(xcb, dtbk, dblb, Alog_b, D_b, xcf, 1);

    // 8) gate by silu(z), then out projection (8192 x 1024 x 512) — WMMA
    k_gate<<<(ROWS * DI + 255) / 256, 256, 0, stream>>>(xcf, xzb);
    gemm(xcf, W_out, hn /*outlin*/, nullptr, ROWS, DM, DI, DI, DI, DM, EP_NONE);

    // 9) final affine combine + h_final
    k_final<<<(ROWS * DM + 255) / 256, 256, 0, stream>>>(hn, dabc, res_w, residp, outp, hfinp);
}